// SelfAttentionLayer_76398878261702
// MI455X (gfx1250) — compile-verified
//
#include <hip/hip_runtime.h>
#include <math.h>

// ---------------------------------------------------------------------------
// Problem constants (fixed by reference setup_inputs)
// ---------------------------------------------------------------------------
constexpr int Bsz = 4;      // batch
constexpr int Cin = 128;    // input channels
constexpr int Nn  = 8192;   // nodes
constexpr int Co  = 256;    // output channels
constexpr int Knb = 32;     // neighbors per node
constexpr int Hh  = 8;      // heads
constexpr int Hd  = 32;     // head dim (Co / Hh)

typedef __attribute__((ext_vector_type(2))) float v2f;
typedef __attribute__((ext_vector_type(8))) float v8f;
typedef __attribute__((ext_vector_type(4))) int   v4i;

#if defined(__HIP_DEVICE_COMPILE__) && __has_builtin(__builtin_amdgcn_global_load_async_to_lds_b128)
#define USE_ASYNC_LDS 1
typedef __attribute__((address_space(1))) v4i gv4i;   // global-AS 16B vector
typedef __attribute__((address_space(3))) v4i sv4i;   // LDS-AS 16B vector
#else
#define USE_ASYNC_LDS 0
#endif

// ---------------------------------------------------------------------------
// Kernel 1/3: fp32 WMMA GEMM   C[M=Co x N=Nn] = A[Co x KK] * B[KK x Nn]
//   A: row-major [Co][KK]  (weights, small, L2-resident)
//   B: BN_NK==false -> stored [KK][Nn]  (x: k-major, n contiguous)
//      BN_NK==true  -> stored [Nn][KK]  (feat: node-major, k contiguous)
//   C: OUT_NK==true  -> stored [Nn][Co] (node-major, for coalesced gathers)
//      OUT_NK==false -> stored [Co][Nn] (reference output layout)
// Block = 128 threads (4 waves), each wave computes a 16(M) x 64(N) strip.
// Per-instruction: V_WMMA_F32_16X16X4_F32, K looped in steps of 4.
// ---------------------------------------------------------------------------
template<int KK, bool BN_NK, bool OUT_NK>
__global__ __launch_bounds__(128)
void gemm_wmma(const float* __restrict__ A,
               const float* __restrict__ Bm,
               float* __restrict__ Cmat)
{
    const int lane = threadIdx.x & 31;
    const int wave = threadIdx.x >> 5;
    const int lm   = lane & 15;       // low-lane index (M or N position)
    const int hi   = lane >> 4;       // 0: K 0/1 (or M 0-7), 1: K 2/3 (M 8-15)

    const int b  = blockIdx.z;
    const int m0 = blockIdx.y * 16;
    const int n0 = blockIdx.x * 256 + wave * 64;

    const float* Bp = Bm   + (size_t)b * KK * Nn;
    float*       Cp = Cmat + (size_t)b * Nn * Co;

    v8f acc0 = {}; v8f acc1 = {}; v8f acc2 = {}; v8f acc3 = {};

    // A fragment base: lane lm holds row m0+lm, K pair (kk + 2*hi, +1)
    const float* Arow = A + (size_t)(m0 + lm) * KK + hi * 2;

    for (int kk = 0; kk < KK; kk += 4) {
        v2f a = *(const v2f*)(Arow + kk);

        v2f b0, b1, b2, b3;
        if (BN_NK) {
            // B stored [n][k]: lane lm holds column n0+..+lm, k contiguous
            const float* Bb = Bp + (size_t)(n0 + lm) * KK + kk + hi * 2;
            b0 = *(const v2f*)(Bb);
            b1 = *(const v2f*)(Bb + 16 * KK);
            b2 = *(const v2f*)(Bb + 32 * KK);
            b3 = *(const v2f*)(Bb + 48 * KK);
        } else {
            // B stored [k][n]: two k-rows, n contiguous across lanes
            const float* Bb = Bp + (size_t)(kk + hi * 2) * Nn + n0 + lm;
            b0.x = Bb[0];      b0.y = Bb[Nn];
            b1.x = Bb[16];     b1.y = Bb[Nn + 16];
            b2.x = Bb[32];     b2.y = Bb[Nn + 32];
            b3.x = Bb[48];     b3.y = Bb[Nn + 48];
        }

        acc0 = __builtin_amdgcn_wmma_f32_16x16x4_f32(false, a, false, b0,
                                                     (short)0, acc0, false, false);
        acc1 = __builtin_amdgcn_wmma_f32_16x16x4_f32(false, a, false, b1,
                                                     (short)0, acc1, false, false);
        acc2 = __builtin_amdgcn_wmma_f32_16x16x4_f32(false, a, false, b2,
                                                     (short)0, acc2, false, false);
        acc3 = __builtin_amdgcn_wmma_f32_16x16x4_f32(false, a, false, b3,
                                                     (short)0, acc3, false, false);
    }

    // C/D layout: lane lm = column n, VGPR r = row m0 + 8*hi + r
    v8f accs[4] = {acc0, acc1, acc2, acc3};
#pragma unroll
    for (int t = 0; t < 4; ++t) {
        const int n = n0 + t * 16 + lm;
        if (OUT_NK) {
            float* base = Cp + (size_t)n * Co + m0 + hi * 8;
#pragma unroll
            for (int r = 0; r < 8; ++r) base[r] = accs[t][r];
        } else {
            float* base = Cp + (size_t)(m0 + hi * 8) * Nn + n;
#pragma unroll
            for (int r = 0; r < 8; ++r) base[(size_t)r * Nn] = accs[t][r];
        }
    }
}

// ---------------------------------------------------------------------------
// Kernel 2: local attention over k=32 neighbors.
// One 256-thread workgroup (8 wave32) per node. Q/K/V/feat are node-major
// [B][N][Co] so every gathered neighbor row is a coalesced 1 KB transfer;
// K+V (67 MB) are L2-resident so the ~2.1 GB of gather traffic stays on-die.
//
// CDNA5 path: GLOBAL_LOAD_ASYNC_TO_LDS_B128 moves the gather straight into
// LDS (ASYNCcnt, no VGPR staging). K and V gathers are both issued up front;
// since async loads complete in order (ISA 08 §4.1), `s_wait_asynccnt 8`
// guarantees the 8 K-pass instructions are done while the 8 V-pass
// instructions continue under the score/softmax compute.
//
// Dynamic LDS: qs(1K) + K(32K) + V(32K) + wsm(1K) + nbr = ~66 KB (<320 KB/WGP).
// Qg and feat may alias: Q row n is consumed (barriered) before feat row n
// is written, and no other block touches row n of either buffer.
// ---------------------------------------------------------------------------
constexpr size_t ATTN_LDS_BYTES =
    (size_t)(Co + 2 * Knb * Co + Hh * Knb) * sizeof(float) + Knb * sizeof(int);

__global__ __launch_bounds__(256)
void attn_kernel(const float* Qg,
                 const float* __restrict__ Kg,
                 const float* __restrict__ Vg,
                 const long long* __restrict__ edges,
                 float* feat)
{
    extern __shared__ char smem[];
    float* qs   = (float*)smem;                 // [Co]
    float* kbuf = qs + Co;                      // [Knb][Co]
    float* vbuf = kbuf + Knb * Co;              // [Knb][Co]
    float* wsm  = vbuf + Knb * Co;              // [Hh][Knb]
    int*   nbr  = (int*)(wsm + Hh * Knb);       // [Knb]

    const int t    = threadIdx.x;
    const int node = blockIdx.x;                // 0 .. B*N-1
    const int b    = node / Nn;
    const int n    = node % Nn;

    if (t < Knb)
        nbr[t] = (int)edges[((size_t)b * Nn + n) * Knb + t];
    qs[t] = Qg[((size_t)b * Nn + n) * Co + t];
    __syncthreads();                            // nbr[] visible

    const float* Kb = Kg + (size_t)b * Nn * Co;
    const float* Vb = Vg + (size_t)b * Nn * Co;

#if USE_ASYNC_LDS
    // each thread owns a 16-byte chunk of 8 rows: j = p*4 + (t>>6)
    const int chunk = (t & 63) * 4;             // float offset of 16B chunk
    const int jbase = t >> 6;
#pragma unroll
    for (int p = 0; p < 8; ++p) {               // K gather: 8 async per wave
        const int j = p * 4 + jbase;
        __builtin_amdgcn_global_load_async_to_lds_b128(
            (gv4i*)(Kb + (size_t)nbr[j] * Co + chunk),
            (sv4i*)(kbuf + j * Co + chunk), 0, 0);
    }
#pragma unroll
    for (int p = 0; p < 8; ++p) {               // V gather: runs under softmax
        const int j = p * 4 + jbase;
        __builtin_amdgcn_global_load_async_to_lds_b128(
            (gv4i*)(Vb + (size_t)nbr[j] * Co + chunk),
            (sv4i*)(vbuf + j * Co + chunk), 0, 0);
    }
    asm volatile("s_wait_asynccnt 0x8" ::: "memory");  // K done (in-order)
    __syncthreads();                            // all waves' K rows in LDS
#else
    for (int j = 0; j < Knb; ++j)
        kbuf[j * Co + t] = Kb[(size_t)nbr[j] * Co + t];
    for (int j = 0; j < Knb; ++j)
        vbuf[j * Co + t] = Vb[(size_t)nbr[j] * Co + t];
    __syncthreads();
#endif

    const int h  = t >> 5;                      // head == wave
    const int lj = t & 31;                      // lane == neighbor

    // scores[h][j] = (1/sqrt(hd)) * dot(q_h, k_{j,h})
    float s = 0.f;
    const float* qh = &qs[h * Hd];
    const float* kr = &kbuf[lj * Co + h * Hd];
#pragma unroll
    for (int d = 0; d < Hd; d += 4)
        s += qh[d] * kr[d] + qh[d + 1] * kr[d + 1]
           + qh[d + 2] * kr[d + 2] + qh[d + 3] * kr[d + 3];
    s *= 0.17677669529663688f;                  // 1/sqrt(32)

    // softmax over 32 neighbors within the wave
    float mx = s;
#pragma unroll
    for (int off = 16; off; off >>= 1) mx = fmaxf(mx, __shfl_xor(mx, off, 32));
    float ex = __expf(s - mx);
    float sum = ex;
#pragma unroll
    for (int off = 16; off; off >>= 1) sum += __shfl_xor(sum, off, 32);
    wsm[h * Knb + lj] = ex / sum;

#if USE_ASYNC_LDS
    asm volatile("s_wait_asynccnt 0x0" ::: "memory");  // V in LDS
#endif
    __syncthreads();                            // V rows + wsm visible

    // att channel c == t (h = t>>5, d = t&31); vbuf[j*Co + t] conflict-free
    float acc = 0.f;
#pragma unroll 8
    for (int j = 0; j < Knb; ++j)
        acc += wsm[h * Knb + j] * vbuf[j * Co + t];

    feat[((size_t)b * Nn + n) * Co + t] = acc;  // coalesced 1 KB row
}

// ---------------------------------------------------------------------------
// Kernel 4: per-channel batch stats over (B, N).  o is [B][Co][N].
// mv[c] = mean, mv[Co + c] = 1/sqrt(var + eps)   (biased var, torch BN)
// ---------------------------------------------------------------------------
__global__ __launch_bounds__(256)
void bn_stats(const float* __restrict__ o, float* __restrict__ mv)
{
    const int c = blockIdx.x;
    float s = 0.f, s2 = 0.f;
    for (int i = threadIdx.x; i < Bsz * Nn; i += 256) {
        const int b = i / Nn, n = i % Nn;
        const float v = o[((size_t)b * Co + c) * Nn + n];
        s += v; s2 += v * v;
    }
    __shared__ float r1[256], r2[256];
    r1[threadIdx.x] = s; r2[threadIdx.x] = s2;
    __syncthreads();
    for (int st = 128; st; st >>= 1) {
        if (threadIdx.x < st) {
            r1[threadIdx.x] += r1[threadIdx.x + st];
            r2[threadIdx.x] += r2[threadIdx.x + st];
        }
        __syncthreads();
    }
    if (threadIdx.x == 0) {
        const float inv  = 1.0f / (float)(Bsz * Nn);
        const float mean = r1[0] * inv;
        const float var  = r2[0] * inv - mean * mean;
        mv[c]      = mean;
        mv[Co + c] = rsqrtf(var + 1e-5f);
    }
}

// ---------------------------------------------------------------------------
// Kernel 5: in-place BN affine + LeakyReLU(0.2) on o = d_out [B][Co][N]
// ---------------------------------------------------------------------------
__global__ __launch_bounds__(256)
void bn_lrelu(float* __restrict__ o, const float* __restrict__ mv,
              const float* __restrict__ gamma, const float* __restrict__ beta)
{
    const size_t idx = (size_t)blockIdx.x * 256 + threadIdx.x;
    const int c = (int)((idx / Nn) % Co);
    float v = o[idx];
    v = (v - mv[c]) * mv[Co + c];
    v = gamma[c] * v + beta[c];
    o[idx] = (v >= 0.f) ? v : 0.2f * v;
}

// ---------------------------------------------------------------------------
// launch
// ---------------------------------------------------------------------------
extern "C" void kernel_launch(void* const* d_in, const int* in_sizes, int n_in,
                              void* d_out, int out_size, void* d_ws, size_t ws_size,
                              hipStream_t stream)
{
    (void)in_sizes; (void)n_in; (void)out_size; (void)ws_size;

    const float*     x     = (const float*)d_in[0];      // [B,Cin,N]
    const long long* edges = (const long long*)d_in[1];  // [B,N,k] int64
    const float*     Wq    = (const float*)d_in[2];      // [Co,Cin]
    const float*     Wk    = (const float*)d_in[3];
    const float*     Wv    = (const float*)d_in[4];
    const float*     Wo    = (const float*)d_in[5];      // [Co,Co]
    const float*     gamma = (const float*)d_in[6];
    const float*     beta  = (const float*)d_in[7];

    float* o = (float*)d_out;                            // [B,Co,N]

    // workspace layout: Q | K | V (node-major [B,N,Co]) | mv (2*Co)
    const size_t mat = (size_t)Bsz * Nn * Co;
    float* Q  = (float*)d_ws;
    float* K  = Q + mat;
    float* V  = K + mat;
    float* mv = V + mat;
    float* feat = Q;   // alias Q: row n is read (and barriered) before written

    // 1) QKV projections: C[Co x Nn] = W * x[b], written node-major
    dim3 g1(Nn / 256, Co / 16, Bsz);
    gemm_wmma<Cin, false, true><<<g1, 128, 0, stream>>>(Wq, x, Q);
    gemm_wmma<Cin, false, true><<<g1, 128, 0, stream>>>(Wk, x, K);
    gemm_wmma<Cin, false, true><<<g1, 128, 0, stream>>>(Wv, x, V);

    // 2) local k-NN attention (one workgroup per node, async LDS gathers)
    attn_kernel<<<Bsz * Nn, 256, ATTN_LDS_BYTES, stream>>>(Q, K, V, edges, feat);

    // 3) output projection: o[b] = Wo * feat[b], reference [Co][Nn] layout
    dim3 g2(Nn / 256, Co / 16, Bsz);
    gemm_wmma<Co, true, false><<<g2, 128, 0, stream>>>(Wo, feat, o);

    // 4) BN batch stats per channel
    bn_stats<<<Co, 256, 0, stream>>>(o, mv);

    // 5) normalize + affine + LeakyReLU, in place on d_out
    const size_t total = (size_t)Bsz * Co * Nn;
    bn_lrelu<<<(unsigned)(total / 256), 256, 0, stream>>>(o, mv, gamma, beta);
}